// Loss_reonstruct_40175124087134
// MI455X (gfx1250) — compile-verified
//
#include <hip/hip_runtime.h>
#include <hip/hip_bf16.h>
#include <math.h>

// ---------------------------------------------------------------------------
// Stereo reconstruction loss (warp + SSIM + LR consistency) for MI455X/gfx1250.
// Phase A: bilinear warps -> reconstructed images in workspace + LR partials.
// Phase B: tiled SSIM; vertical 3-tap box sums done on the matrix pipe via
//          V_WMMA_F32_16X16X4_F32 (banded-matrix x tile), wave32 per tile.
// Phase C: deterministic single-block reduction -> 3 scalar outputs.
// Workspace requirement: 2*N*C*H*W + 20480 + 2*61440 floats  (~126.4 MB).
// ---------------------------------------------------------------------------

#define NB 4
#define CC 3
#define HH 1024
#define WW 1280

static constexpr float S_C1 = 0.01f * 0.01f;
static constexpr float S_C2 = 0.03f * 0.03f;

typedef __attribute__((ext_vector_type(2))) float v2f;
typedef __attribute__((ext_vector_type(8))) float v8f;

// ---------------------------------------------------------------------------
// Bilinear sample with zeros padding. Vertical coordinate is always i-0.5
// (rows i-1 and i, weights 0.5/0.5), derived exactly from the reference grid.
// ---------------------------------------------------------------------------
__device__ __forceinline__ float sample_bilinear(const float* __restrict__ plane,
                                                 int i, int ix0, float fx,
                                                 bool vx0, bool vx1)
{
    const int  y0  = i - 1;
    const bool y0v = (y0 >= 0);
    const int  yc0 = y0v ? y0 : 0;
    const int  xc0 = vx0 ? ix0 : 0;
    const int  xc1 = vx1 ? (ix0 + 1) : 0;
    const float* r0 = plane + (size_t)yc0 * WW;
    const float* r1 = plane + (size_t)i   * WW;
    const float m00 = (y0v && vx0) ? 1.0f : 0.0f;
    const float m10 = vx0 ? 1.0f : 0.0f;
    const float m01 = (y0v && vx1) ? 1.0f : 0.0f;
    const float m11 = vx1 ? 1.0f : 0.0f;
    const float Ia = r0[xc0] * m00;   // (x0, y0)
    const float Ib = r1[xc0] * m10;   // (x0, y1)
    const float Ic = r0[xc1] * m01;   // (x1, y0)
    const float Id = r1[xc1] * m11;   // (x1, y1)
    return 0.5f * ((1.0f - fx) * (Ia + Ib) + fx * (Ic + Id));
}

// ---------------------------------------------------------------------------
// Phase A: one thread per (n,i,j). Exact cover: N*H*W == 20480*256.
// ---------------------------------------------------------------------------
__global__ __launch_bounds__(256) void warp_lr_kernel(
    const float* __restrict__ left, const float* __restrict__ right,
    const float* __restrict__ ldisp, const float* __restrict__ rdisp,
    float* __restrict__ lrec, float* __restrict__ rrec,
    float* __restrict__ partial_lr)
{
    const int gid = blockIdx.x * 256 + threadIdx.x;
    const int j = gid % WW;
    const int t = gid / WW;
    const int i = t % HH;
    const int n = t / HH;

    const float dl = ldisp[gid];
    const float dr = rdisp[gid];
    const float gx = -1.0f + (float)j * (2.0f / (float)WW);

    // x = (gx + sign*disp + 1) * W/2 - 0.5   (replicate reference op order)
    const float xl  = ((gx - dl) + 1.0f) * ((float)WW * 0.5f) - 0.5f;
    const float xl0 = floorf(xl);
    const float fxl = xl - xl0;
    const int   ixl = (int)xl0;
    const bool  vl0 = (xl0 >= 0.0f) && (xl0 < (float)WW);
    const bool  vl1 = (xl0 + 1.0f >= 0.0f) && (xl0 + 1.0f < (float)WW);

    const float xr  = ((gx + dr) + 1.0f) * ((float)WW * 0.5f) - 0.5f;
    const float xr0 = floorf(xr);
    const float fxr = xr - xr0;
    const int   ixr = (int)xr0;
    const bool  vr0 = (xr0 >= 0.0f) && (xr0 < (float)WW);
    const bool  vr1 = (xr0 + 1.0f >= 0.0f) && (xr0 + 1.0f < (float)WW);

    const size_t pstride = (size_t)HH * WW;
    const size_t pix     = (size_t)i * WW + j;

#pragma unroll
    for (int c = 0; c < CC; ++c) {
        const size_t pbase = (size_t)(n * CC + c) * pstride;
        lrec[pbase + pix] = sample_bilinear(right + pbase, i, ixl, fxl, vl0, vl1);
        rrec[pbase + pix] = sample_bilinear(left  + pbase, i, ixr, fxr, vr0, vr1);
    }

    // LR consistency: warp right_disp with the left-view coordinates.
    const float rw = sample_bilinear(rdisp + (size_t)n * pstride, i, ixl, fxl, vl0, vl1);
    const float ad = fabsf(dl - rw);

    __shared__ float red[256];
    red[threadIdx.x] = ad;
    __syncthreads();
#pragma unroll
    for (int off = 128; off > 0; off >>= 1) {
        if (threadIdx.x < off) red[threadIdx.x] += red[threadIdx.x + off];
        __syncthreads();
    }
    if (threadIdx.x == 0) partial_lr[blockIdx.x] = red[0];
}

// ---------------------------------------------------------------------------
// Phase B: SSIM. One wave32 per 16x16 output tile. Vertical 3-tap pooling via
// WMMA: P(16x16) = S(16x20, banded) * H(20x16), K-chunked into 5 x (K=4).
// ---------------------------------------------------------------------------
#define TI 64                    // ceil(1022/16)
#define TJ 80                    // ceil(1278/16)
#define TILES (TI * TJ)          // 5120
#define PER_PAIR (NB * CC * TILES)  // 61440

__global__ __launch_bounds__(32) void ssim_kernel(
    const float* __restrict__ lrec, const float* __restrict__ rrec,
    const float* __restrict__ left, const float* __restrict__ right,
    float* __restrict__ partial)   // [2 * PER_PAIR]
{
    const int bx   = blockIdx.x;
    const int pair = bx / PER_PAIR;
    const int r    = bx % PER_PAIR;
    const int nc   = r / TILES;
    const int tile = r % TILES;
    const int ti   = tile / TJ;
    const int tj   = tile % TJ;
    const int i0   = ti * 16;
    const int j0   = tj * 16;

    const size_t pbase = (size_t)nc * HH * WW;
    const float* X = (pair == 0 ? lrec : rrec) + pbase;   // reconstruction
    const float* Y = (pair == 0 ? left : right) + pbase;  // original

    __shared__ float xt[18][19];
    __shared__ float yt[18][19];
    __shared__ float Hs[5][20][16];   // horizontal 3-tap sums; rows 18-19 zero

    const int l = threadIdx.x;

    // Stage 18x18 input tiles (clamped loads, zero-masked OOB).
    for (int t = l; t < 18 * 18; t += 32) {
        const int rr = t / 18, cc = t % 18;
        const int gi = i0 + rr, gj = j0 + cc;
        const bool v = (gi < HH) && (gj < WW);
        const int  gic = v ? gi : (HH - 1);
        const int  gjc = v ? gj : (WW - 1);
        const size_t idx = (size_t)gic * WW + gjc;
        const float m = v ? 1.0f : 0.0f;
        xt[rr][cc] = X[idx] * m;
        yt[rr][cc] = Y[idx] * m;
    }
    __syncthreads();

    // Horizontal 3-tap sums of {x, y, x^2, y^2, xy}; zero K-pad rows 18..19.
    for (int t = l; t < 20 * 16; t += 32) {
        const int rr = t / 16, cc = t % 16;
        float sx = 0.f, sy = 0.f, sxx = 0.f, syy = 0.f, sxy = 0.f;
        if (rr < 18) {
#pragma unroll
            for (int d = 0; d < 3; ++d) {
                const float xv = xt[rr][cc + d];
                const float yv = yt[rr][cc + d];
                sx += xv; sy += yv;
                sxx += xv * xv; syy += yv * yv; sxy += xv * yv;
            }
        }
        Hs[0][rr][cc] = sx;  Hs[1][rr][cc] = sy;
        Hs[2][rr][cc] = sxx; Hs[3][rr][cc] = syy; Hs[4][rr][cc] = sxy;
    }
    __syncthreads();

    // WMMA accumulation on the matrix pipe.
    // A 16x4 f32 layout: lanes0-15 M=0..15 {K,K+1 in .x/.y? no: VGPR0 = K,K+2}:
    //   VGPR0: K=k0 (lanes0-15), K=k0+2 (lanes16-31); VGPR1: K=k0+1 / K=k0+3.
    // B 4x16 mirrors with N along lanes.
    const int m  = l & 15;            // A: row this lane holds
    const int nn = l & 15;            // B/C/D: column this lane holds
    const int kb = (l >> 4) * 2;      // lane-half K offset

    v8f acc[5] = {};
#pragma unroll
    for (int q = 0; q < 5; ++q) {
        const int k0 = 4 * q + kb;
        v2f a;
        a.x = (k0     >= m && k0     <= m + 2) ? 1.0f : 0.0f;
        a.y = (k0 + 1 >= m && k0 + 1 <= m + 2) ? 1.0f : 0.0f;
#pragma unroll
        for (int s = 0; s < 5; ++s) {
            v2f b;
            b.x = Hs[s][k0][nn];
            b.y = Hs[s][k0 + 1][nn];
            acc[s] = __builtin_amdgcn_wmma_f32_16x16x4_f32(
                /*neg_a=*/false, a, /*neg_b=*/false, b,
                /*c_mod=*/(short)0, acc[s],
                /*reuse_a=*/false, /*reuse_b=*/false);
        }
    }

    // Each lane owns 8 output pixels: M = (l>=16 ? 8 : 0) + g, N = l&15.
    const int mrow = (l >> 4) * 8;
    const float inv9 = 1.0f / 9.0f;
    float lsum = 0.0f;
#pragma unroll
    for (int g = 0; g < 8; ++g) {
        const int oi = i0 + mrow + g;
        const int oj = j0 + nn;
        const float mux = acc[0][g] * inv9;
        const float muy = acc[1][g] * inv9;
        const float sgx = acc[2][g] * inv9 - mux * mux;
        const float sgy = acc[3][g] * inv9 - muy * muy;
        const float sxy = acc[4][g] * inv9 - mux * muy;
        const float nmr = (2.0f * mux * muy + S_C1) * (2.0f * sxy + S_C2);
        const float dnm = (mux * mux + muy * muy + S_C1) * (sgx + sgy + S_C2);
        float v = (1.0f - nmr / dnm) * 0.5f;
        v = fminf(fmaxf(v, 0.0f), 1.0f);
        const bool valid = (oi < HH - 2) && (oj < WW - 2);
        lsum += valid ? v : 0.0f;
    }

    // Deterministic wave32 reduction.
#pragma unroll
    for (int off = 16; off > 0; off >>= 1)
        lsum += __shfl_xor(lsum, off, 32);
    if (l == 0) partial[bx] = lsum;
}

// ---------------------------------------------------------------------------
// Phase C: deterministic single-block reduction of all partials -> 3 scalars.
// ---------------------------------------------------------------------------
#define NLR ((NB * HH * WW) / 256)   // 20480

__global__ __launch_bounds__(256) void finalize_kernel(
    const float* __restrict__ p_lr,
    const float* __restrict__ p_ss,   // [2][PER_PAIR]
    float* __restrict__ out)
{
    __shared__ float r0[256], r1[256], r2[256];
    const int t = threadIdx.x;
    float a = 0.f, b = 0.f, c = 0.f;
    for (int k = t; k < PER_PAIR; k += 256) a += p_ss[k];
    for (int k = t; k < PER_PAIR; k += 256) b += p_ss[PER_PAIR + k];
    for (int k = t; k < NLR; k += 256)      c += p_lr[k];
    r0[t] = a; r1[t] = b; r2[t] = c;
    __syncthreads();
#pragma unroll
    for (int off = 128; off > 0; off >>= 1) {
        if (t < off) { r0[t] += r0[t + off]; r1[t] += r1[t + off]; r2[t] += r2[t + off]; }
        __syncthreads();
    }
    if (t == 0) {
        const float cntS = (float)NB * CC * (HH - 2) * (WW - 2);
        const float cntL = (float)NB * HH * WW;
        out[0] = r0[0] / cntS;
        out[1] = r1[0] / cntS;
        out[2] = r2[0] / cntL;
    }
}

// ---------------------------------------------------------------------------
extern "C" void kernel_launch(void* const* d_in, const int* in_sizes, int n_in,
                              void* d_out, int out_size, void* d_ws, size_t ws_size,
                              hipStream_t stream)
{
    (void)in_sizes; (void)n_in; (void)out_size; (void)ws_size;

    const float* left  = (const float*)d_in[0];
    const float* right = (const float*)d_in[1];
    const float* ldisp = (const float*)d_in[2];
    const float* rdisp = (const float*)d_in[3];
    float* out = (float*)d_out;

    const size_t plane_total = (size_t)NB * CC * HH * WW;  // 15,728,640 floats
    float* ws    = (float*)d_ws;
    float* lrec  = ws;
    float* rrec  = lrec + plane_total;
    float* p_lr  = rrec + plane_total;        // NLR floats
    float* p_ss  = p_lr + NLR;                // 2 * PER_PAIR floats

    warp_lr_kernel<<<(NB * HH * WW) / 256, 256, 0, stream>>>(
        left, right, ldisp, rdisp, lrec, rrec, p_lr);

    ssim_kernel<<<2 * PER_PAIR, 32, 0, stream>>>(
        lrec, rrec, left, right, p_ss);

    finalize_kernel<<<1, 256, 0, stream>>>(p_lr, p_ss, out);
}